// Net_34024730919296
// MI455X (gfx1250) — compile-verified
//
#include <hip/hip_runtime.h>
#include <cstdint>
#include <cstddef>

// ---------------------------------------------------------------------------
// LUT-network (3 layers, SIX=6, LUT=64, B=IN=F=2048), fused per batch row.
//
// Roofline: unique HBM traffic ~= r1..r3 (288 MB, streamed NT) + inputs
// (16 MB) + out (16 MB) => ~14 us floor at 23.3 TB/s.  No matmul in this
// workload -> no WMMA; CDNA5 paths used: async global->LDS DMA (ASYNCcnt),
// LDS ds gathers/stores, NT temporal hints, split barriers.
//
// v2: cut instruction-issue pressure (same HBM bytes):
//   - feature pairs  -> r loads become 3x NT b128 per pair (1.5 vmem/feat)
//   - connect packed -> u16 byte-offsets in d_ws (1.5 vmem/feat vs 6)
//   - paired ds_store_b64 + NT b64 output stores
// ---------------------------------------------------------------------------

constexpr int kSIX     = 6;
constexpr int kLUT     = 64;
constexpr int kB       = 2048;
constexpr int kIN      = 2048;
constexpr int kF       = 2048;
constexpr int kThreads = 256;
constexpr int kFPT     = kF / kThreads; // 8 features per thread per layer

typedef __attribute__((ext_vector_type(2))) float v2f;
typedef __attribute__((ext_vector_type(4))) float v4f;

__device__ __forceinline__ float fast_sigmoid(float v) {
  return __builtin_amdgcn_rcpf(1.0f + __expf(-v)); // v_exp_f32 + v_rcp_f32
}

__device__ __forceinline__ void wait_asynccnt0() {
#if __has_builtin(__builtin_amdgcn_s_wait_asynccnt)
  __builtin_amdgcn_s_wait_asynccnt(0);
#else
  asm volatile("s_wait_asynccnt 0" ::: "memory");
#endif
}

// Stage one 8 KB row from global into LDS via gfx1250 async DMA:
// 256 threads x b128 x 2 passes.  LDS offset = low 32 bits of the generic
// shared pointer (aperture layout: addr[31:0] == LDS byte offset).
__device__ __forceinline__ void async_copy_row(const float* __restrict__ gsrc,
                                               float* lds_dst, int tid) {
  uint32_t lbase = (uint32_t)(uintptr_t)lds_dst;
  uint64_t gbase = (uint64_t)(uintptr_t)gsrc;
#pragma unroll
  for (int it = 0; it < 2; ++it) {
    uint32_t off = (uint32_t)tid * 16u + (uint32_t)it * (kThreads * 16u);
    uint32_t l = lbase + off;
    uint64_t g = gbase + off;
    asm volatile("global_load_async_to_lds_b128 %0, %1, off"
                 :: "v"(l), "v"(g) : "memory");
  }
}

// ---------------------------------------------------------------------------
// Prep: repack int64 connect indices into u16 BYTE offsets (idx*4 <= 8188).
// 12 B/feature instead of 48 B; feeds ds_load addressing with no shift.
// ---------------------------------------------------------------------------
__global__ __launch_bounds__(kThreads)
void pack_connect_kernel(const long long* __restrict__ c1,
                         const long long* __restrict__ c2,
                         const long long* __restrict__ c3,
                         unsigned short* __restrict__ p1,
                         unsigned short* __restrict__ p2,
                         unsigned short* __restrict__ p3) {
  int i = blockIdx.x * blockDim.x + threadIdx.x;
  if (i < kF * kSIX) {
    p1[i] = (unsigned short)(((unsigned)c1[i]) << 2);
    p2[i] = (unsigned short)(((unsigned)c2[i]) << 2);
    p3[i] = (unsigned short)(((unsigned)c3[i]) << 2);
  }
}

// ---------------------------------------------------------------------------
// Packed-connect layer: thread handles feature pair (f0, f0+1).
// ---------------------------------------------------------------------------
template <bool LAST>
__device__ __forceinline__ void lut_layer_packed(
    const float* __restrict__ xin,                 // LDS row (prev layer)
    float* __restrict__ xout,                      // LDS row (if !LAST)
    const float* __restrict__ r_row,               // r + b*F*6
    const float* __restrict__ lut,                 // [F,64], L2-hot
    const unsigned short* __restrict__ conp,       // [F*6] u16 byte-offsets
    float* __restrict__ gout_row,                  // out + b*F (if LAST)
    int tid) {
#pragma unroll
  for (int k = 0; k < kFPT / 2; ++k) {
    const int p  = tid + k * kThreads;             // lane-coalesced pair id
    const int f0 = 2 * p;
    // 48 B of uniforms per pair, 16-aligned, streamed once -> 3x NT b128
    const v4f* rp = (const v4f*)(r_row + (size_t)f0 * kSIX);
    v4f ra = __builtin_nontemporal_load(rp + 0);   // f0: r0..r3
    v4f rb = __builtin_nontemporal_load(rp + 1);   // f0: r4,r5 | f1: r0,r1
    v4f rc = __builtin_nontemporal_load(rp + 2);   // f1: r2..r5
    // 24 B of packed offsets per pair (L2-resident) -> 3x b64
    const uint64_t* cp = (const uint64_t*)(conp + (size_t)p * 12);
    uint64_t q0 = cp[0], q1 = cp[1], q2 = cp[2];
    uint32_t a0 = (uint32_t)(q0      ) & 0xFFFFu;
    uint32_t a1 = (uint32_t)(q0 >> 16) & 0xFFFFu;
    uint32_t a2 = (uint32_t)(q0 >> 32) & 0xFFFFu;
    uint32_t a3 = (uint32_t)(q0 >> 48);
    uint32_t a4 = (uint32_t)(q1      ) & 0xFFFFu;
    uint32_t a5 = (uint32_t)(q1 >> 16) & 0xFFFFu;
    uint32_t b0 = (uint32_t)(q1 >> 32) & 0xFFFFu;
    uint32_t b1 = (uint32_t)(q1 >> 48);
    uint32_t b2 = (uint32_t)(q2      ) & 0xFFFFu;
    uint32_t b3 = (uint32_t)(q2 >> 16) & 0xFFFFu;
    uint32_t b4 = (uint32_t)(q2 >> 32) & 0xFFFFu;
    uint32_t b5 = (uint32_t)(q2 >> 48);
    const char* xc = (const char*)xin;
    // 12 random ds_load_b32 gathers from the 8 KB LDS row
    float x0 = *(const float*)(xc + a0);
    float x1 = *(const float*)(xc + a1);
    float x2 = *(const float*)(xc + a2);
    float x3 = *(const float*)(xc + a3);
    float x4 = *(const float*)(xc + a4);
    float x5 = *(const float*)(xc + a5);
    float y0 = *(const float*)(xc + b0);
    float y1 = *(const float*)(xc + b1);
    float y2 = *(const float*)(xc + b2);
    float y3 = *(const float*)(xc + b3);
    float y4 = *(const float*)(xc + b4);
    float y5 = *(const float*)(xc + b5);
    int idx0 = (x0 >= ra.x ?  1 : 0) | (x1 >= ra.y ?  2 : 0)
             | (x2 >= ra.z ?  4 : 0) | (x3 >= ra.w ?  8 : 0)
             | (x4 >= rb.x ? 16 : 0) | (x5 >= rb.y ? 32 : 0);
    int idx1 = (y0 >= rb.z ?  1 : 0) | (y1 >= rb.w ?  2 : 0)
             | (y2 >= rc.x ?  4 : 0) | (y3 >= rc.y ?  8 : 0)
             | (y4 >= rc.z ? 16 : 0) | (y5 >= rc.w ? 32 : 0);
    float v0 = lut[f0 * kLUT + idx0];              // L2-resident lookups
    float v1 = lut[f0 * kLUT + kLUT + idx1];
    if (LAST) {
      v2f o; o.x = v0; o.y = v1;
      __builtin_nontemporal_store(o, (v2f*)(gout_row + f0)); // NT b64 store
    } else {
      v2f s; s.x = fast_sigmoid(v0); s.y = fast_sigmoid(v1);
      *(v2f*)(xout + f0) = s;                      // one ds_store_b64
    }
  }
}

__global__ __launch_bounds__(kThreads)
void lutnet_fused_packed(const float* __restrict__ inputs,
                         const float* __restrict__ r1,
                         const float* __restrict__ r2,
                         const float* __restrict__ r3,
                         const float* __restrict__ lut1,
                         const float* __restrict__ lut2,
                         const float* __restrict__ lut3,
                         const unsigned short* __restrict__ p1,
                         const unsigned short* __restrict__ p2,
                         const unsigned short* __restrict__ p3,
                         float* __restrict__ out) {
  __shared__ float xa[kIN];   // 8 KB ping
  __shared__ float xb[kF];    // 8 KB pong
  const int b = blockIdx.x;
  const int t = threadIdx.x;

  async_copy_row(inputs + (size_t)b * kIN, xa, t);
  wait_asynccnt0();
  __syncthreads();
#pragma unroll
  for (int k = 0; k < kFPT; ++k) {
    int i = t + k * kThreads;
    xa[i] = (xa[i] + 1.0f) * 0.5f;
  }
  __syncthreads();

  const size_t rstride = (size_t)kF * kSIX;
  lut_layer_packed<false>(xa, xb, r1 + (size_t)b * rstride, lut1, p1, nullptr, t);
  __syncthreads();
  lut_layer_packed<false>(xb, xa, r2 + (size_t)b * rstride, lut2, p2, nullptr, t);
  __syncthreads();
  lut_layer_packed<true >(xa, nullptr, r3 + (size_t)b * rstride, lut3, p3,
                          out + (size_t)b * kF, t);
}

// ---------------------------------------------------------------------------
// Fallback (round-1 path, int64 connect straight from global) in case
// the workspace is too small for the packed-connect tables.
// ---------------------------------------------------------------------------
template <bool LAST>
__device__ __forceinline__ void lut_layer_raw(
    const float* __restrict__ xin, float* __restrict__ xout,
    const float* __restrict__ r_row, const float* __restrict__ lut,
    const long long* __restrict__ con, float* __restrict__ gout_row, int tid) {
#pragma unroll
  for (int k = 0; k < kFPT; ++k) {
    const int f = tid + k * kThreads;
    const v2f* rp = (const v2f*)(r_row + (size_t)f * kSIX);
    v2f r01 = __builtin_nontemporal_load(rp + 0);
    v2f r23 = __builtin_nontemporal_load(rp + 1);
    v2f r45 = __builtin_nontemporal_load(rp + 2);
    const long long* cp = con + (size_t)f * kSIX;
    int i0 = (int)cp[0], i1 = (int)cp[1], i2 = (int)cp[2];
    int i3 = (int)cp[3], i4 = (int)cp[4], i5 = (int)cp[5];
    float x0 = xin[i0], x1 = xin[i1], x2 = xin[i2];
    float x3 = xin[i3], x4 = xin[i4], x5 = xin[i5];
    int idx = (x0 >= r01.x ?  1 : 0) | (x1 >= r01.y ?  2 : 0)
            | (x2 >= r23.x ?  4 : 0) | (x3 >= r23.y ?  8 : 0)
            | (x4 >= r45.x ? 16 : 0) | (x5 >= r45.y ? 32 : 0);
    float v = lut[f * kLUT + idx];
    if (LAST) __builtin_nontemporal_store(v, gout_row + f);
    else      xout[f] = fast_sigmoid(v);
  }
}

__global__ __launch_bounds__(kThreads)
void lutnet_fused_raw(const float* __restrict__ inputs,
                      const float* __restrict__ r1,
                      const float* __restrict__ r2,
                      const float* __restrict__ r3,
                      const float* __restrict__ lut1,
                      const float* __restrict__ lut2,
                      const float* __restrict__ lut3,
                      const long long* __restrict__ c1,
                      const long long* __restrict__ c2,
                      const long long* __restrict__ c3,
                      float* __restrict__ out) {
  __shared__ float xa[kIN];
  __shared__ float xb[kF];
  const int b = blockIdx.x;
  const int t = threadIdx.x;

  async_copy_row(inputs + (size_t)b * kIN, xa, t);
  wait_asynccnt0();
  __syncthreads();
#pragma unroll
  for (int k = 0; k < kFPT; ++k) {
    int i = t + k * kThreads;
    xa[i] = (xa[i] + 1.0f) * 0.5f;
  }
  __syncthreads();

  const size_t rstride = (size_t)kF * kSIX;
  lut_layer_raw<false>(xa, xb, r1 + (size_t)b * rstride, lut1, c1, nullptr, t);
  __syncthreads();
  lut_layer_raw<false>(xb, xa, r2 + (size_t)b * rstride, lut2, c2, nullptr, t);
  __syncthreads();
  lut_layer_raw<true >(xa, nullptr, r3 + (size_t)b * rstride, lut3, c3,
                       out + (size_t)b * kF, t);
}

extern "C" void kernel_launch(void* const* d_in, const int* in_sizes, int n_in,
                              void* d_out, int out_size, void* d_ws, size_t ws_size,
                              hipStream_t stream) {
  (void)in_sizes; (void)n_in; (void)out_size;
  const float*     inputs = (const float*)d_in[0];
  const float*     r1     = (const float*)d_in[1];
  const float*     r2     = (const float*)d_in[2];
  const float*     r3     = (const float*)d_in[3];
  const float*     lut1   = (const float*)d_in[4];
  const float*     lut2   = (const float*)d_in[5];
  const float*     lut3   = (const float*)d_in[6];
  const long long* c1     = (const long long*)d_in[7];
  const long long* c2     = (const long long*)d_in[8];
  const long long* c3     = (const long long*)d_in[9];
  float* out = (float*)d_out;

  const size_t tblElems = (size_t)kF * kSIX;           // 12288
  const size_t tblBytes = tblElems * sizeof(unsigned short);
  if (ws_size >= 3 * tblBytes) {
    unsigned short* p1 = (unsigned short*)d_ws;
    unsigned short* p2 = p1 + tblElems;
    unsigned short* p3 = p2 + tblElems;
    const int prepBlocks = (int)((tblElems + kThreads - 1) / kThreads);
    pack_connect_kernel<<<prepBlocks, kThreads, 0, stream>>>(c1, c2, c3,
                                                             p1, p2, p3);
    lutnet_fused_packed<<<kB, kThreads, 0, stream>>>(
        inputs, r1, r2, r3, lut1, lut2, lut3, p1, p2, p3, out);
  } else {
    lutnet_fused_raw<<<kB, kThreads, 0, stream>>>(
        inputs, r1, r2, r3, lut1, lut2, lut3, c1, c2, c3, out);
  }
}